// VoxelBackBone8xFusion_28381143892901
// MI455X (gfx1250) — compile-verified
//
#include <hip/hip_runtime.h>

typedef __attribute__((ext_vector_type(2))) float v2f;
typedef __attribute__((ext_vector_type(4))) float v4f;
typedef __attribute__((ext_vector_type(8))) float v8f;
typedef __attribute__((ext_vector_type(4))) int   v4i;

#define N_VOX  1000000
#define C_PTS  16
#define C_IMG  16
#define IMG_H  376
#define IMG_W  1248

// Each wave processes tiles of 16 voxels.
//  - Projection einsum done with V_WMMA_F32_16X16X4_F32:
//      A[m,k]      = hom coords of voxel m (16x4)
//      B[k,b*3+j]  = calib_P[b][j][k]   (all 4 batches packed into 12 cols)
//      D[m,b*3+j]  = projection of voxel m under batch b
//  - D scattered to a per-wave LDS slab (stride 17, bank-conflict free),
//    each voxel picks its batch's 3 columns.
//  - Lanes l and l+16 co-own voxel (l&15): each half handles 8 image channels
//    and 8 voxel-feature floats -> b128 loads/stores, no divergence.
//  - Cache policy: x_rgb (120 MB) must stay resident in the 192 MB L2 so the
//    16M random gathers hit; all touch-once streams (indices, voxel features,
//    output) use non-temporal hints so they don't evict it.
__global__ __launch_bounds__(256) void voxfuse_kernel(
    const float* __restrict__ vox_feat,   // [N_VOX, 16]
    const float* __restrict__ x_rgb,      // [4, 16, H, W]
    const float* __restrict__ calibP,     // [4, 3, 4]
    const int*   __restrict__ vox_idx,    // [N_VOX, 4] (b, z, y, x)
    float* __restrict__ out)              // [N_VOX, 32]
{
    __shared__ float lds[8][16 * 17 + 4];   // per-wave proj scratch

    const int lane = threadIdx.x & 31;
    const int wave = threadIdx.x >> 5;
    const int col  = lane & 15;
    const int hi   = lane >> 4;             // 0: K={0,1}, 1: K={2,3}

    // ---- B matrix, built once: 4x16 f32. Lanes 0-15 hold K=0/1, 16-31 K=2/3.
    v2f bmat;
    bmat.x = 0.0f; bmat.y = 0.0f;
    if (col < 12) {
        const int b_ = col / 3;
        const int j  = col % 3;
        const float* p = calibP + b_ * 12 + j * 4;
        bmat.x = hi ? p[2] : p[0];
        bmat.y = hi ? p[3] : p[1];
    }

    const int numTiles   = (N_VOX + 15) >> 4;               // 62500
    const int wavesTotal = (gridDim.x * blockDim.x) >> 5;
    int waveId = (blockIdx.x * blockDim.x + threadIdx.x) >> 5;

    const size_t planeStride = (size_t)IMG_H * IMG_W;
    float* const lw = &lds[wave][0];

    for (int tile = waveId; tile < numTiles; tile += wavesTotal) {
        const int base = tile << 4;
        const int n    = base + col;                         // voxel this lane serves
        const int nc   = (n < N_VOX) ? n : (N_VOX - 1);

        // Prefetch next tile's streaming inputs (global_prefetch_b8).
        const int nt = tile + wavesTotal;
        if (nt < numTiles) {
            __builtin_prefetch(vox_feat + (size_t)(nt << 4) * C_PTS, 0, 1);
            __builtin_prefetch(vox_idx  + (size_t)(nt << 4) * 4,     0, 1);
        }

        // ---- Load indices (NT: touch-once stream), build homogeneous point.
        const v4i vi   = __builtin_nontemporal_load((const v4i*)(vox_idx + (size_t)nc * 4));
        const int bidx = vi.x;
        const float zc = (float)vi.y * 0.10f - 3.0f;
        const float yc = (float)vi.z * 0.05f - 40.0f;
        const float xc = (float)vi.w * 0.05f + 0.0f;         // hom = (xc, yc, zc, 1)

        // ---- A matrix 16x4 f32: VGPR0 = K0 (lo) / K2 (hi), VGPR1 = K1 / K3.
        v2f amat;
        amat.x = hi ? zc   : xc;
        amat.y = hi ? 1.0f : yc;

        v8f acc = {0.f, 0.f, 0.f, 0.f, 0.f, 0.f, 0.f, 0.f};
        acc = __builtin_amdgcn_wmma_f32_16x16x4_f32(
                  /*neg_a=*/false, amat, /*neg_b=*/false, bmat,
                  /*c_mod=*/(short)0, acc, /*reuse_a=*/false, /*reuse_b=*/false);

        // ---- Scatter D to LDS: VGPR r of lane holds D[m = r + 8*hi][col].
        {
            const int mb = 8 * hi;
            #pragma unroll
            for (int r = 0; r < 8; ++r)
                lw[col * 17 + (mb + r)] = acc[r];
        }
        __asm volatile("s_wait_dscnt 0" ::: "memory");  // wave32 lockstep: LDS now visible

        // ---- Pick this voxel's batch columns, perspective divide.
        const int c0   = bidx * 3;
        const float px = lw[(c0 + 0) * 17 + col];
        const float py = lw[(c0 + 1) * 17 + col];
        const float pz = lw[(c0 + 2) * 17 + col];

        // Shared raw v_rcp_f32 (<=1 ulp): result feeds an int truncation, and
        // rcp preserves pz->0 => +/-inf => saturating cvt => invalid semantics.
        const float rz = __builtin_amdgcn_rcpf(pz);
        const float uf = px * rz;
        const float vf = py * rz;
        const int ui  = (int)uf;      // v_cvt_i32_f32: truncate, saturate, NaN->0
        const int vti = (int)vf;
        const bool valid = (ui >= 0) & (ui < IMG_W) & (vti >= 0) & (vti < IMG_H);
        const int uc = min(max(ui, 0), IMG_W - 1);
        const int vc = min(max(vti, 0), IMG_H - 1);

        // ---- Gather 8 image channels (RT policy: keep x_rgb hot in L2).
        const float* ib = x_rgb + ((size_t)bidx * C_IMG + (size_t)hi * 8) * planeStride
                                + (size_t)vc * IMG_W + (size_t)uc;
        v4f img0, img1;
        img0.x = ib[0 * planeStride];
        img0.y = ib[1 * planeStride];
        img0.z = ib[2 * planeStride];
        img0.w = ib[3 * planeStride];
        img1.x = ib[4 * planeStride];
        img1.y = ib[5 * planeStride];
        img1.z = ib[6 * planeStride];
        img1.w = ib[7 * planeStride];
        if (!valid) {
            img0 = (v4f){0.f, 0.f, 0.f, 0.f};
            img1 = (v4f){0.f, 0.f, 0.f, 0.f};
        }

        // ---- Streaming voxel features (NT b128 loads), this half's 8 floats.
        const v4f* vfp = (const v4f*)(vox_feat + (size_t)nc * C_PTS + (size_t)hi * 8);
        const v4f vf0 = __builtin_nontemporal_load(vfp + 0);
        const v4f vf1 = __builtin_nontemporal_load(vfp + 1);

        // ---- Store out[n, :] = concat(img 16, voxfeat 16); NT b128 stores so
        //      the 128 MB output stream doesn't evict x_rgb from L2.
        if (n < N_VOX) {
            v4f* o  = (v4f*)(out + (size_t)n * 32 + (size_t)hi * 8);
            __builtin_nontemporal_store(img0, o + 0);
            __builtin_nontemporal_store(img1, o + 1);
            v4f* o2 = (v4f*)(out + (size_t)n * 32 + 16 + (size_t)hi * 8);
            __builtin_nontemporal_store(vf0, o2 + 0);
            __builtin_nontemporal_store(vf1, o2 + 1);
        }
    }
}

extern "C" void kernel_launch(void* const* d_in, const int* in_sizes, int n_in,
                              void* d_out, int out_size, void* d_ws, size_t ws_size,
                              hipStream_t stream) {
    const float* vox_feat = (const float*)d_in[0];   // voxel_features [1e6,16] f32
    const float* x_rgb    = (const float*)d_in[1];   // [4,16,376,1248] f32
    const float* calibP   = (const float*)d_in[2];   // [4,3,4] f32
    const int*   vox_idx  = (const int*)d_in[3];     // [1e6,4] i32
    float* outp = (float*)d_out;                     // [1e6,32] f32

    dim3 grid(2048);
    dim3 block(256);   // 8 waves/block on wave32
    hipLaunchKernelGGL(voxfuse_kernel, grid, block, 0, stream,
                       vox_feat, x_rgb, calibP, vox_idx, outp);
}